// GHTGraphBuilder_11553462026731
// MI455X (gfx1250) — compile-verified
//
#include <hip/hip_runtime.h>
#include <math.h>

#define BATCH 32
#define NTOK  1024
#define DIM   256
#define KNN   8

typedef __attribute__((ext_vector_type(2))) float v2f;
typedef __attribute__((ext_vector_type(8))) float v8f;

// ---------------------------------------------------------------------------
// Kernel 1: per-token reciprocal norm  rinv[b*N+n] = 1/(||t||+1e-8)
// One wave (32 lanes) per token row of 256 floats.
// ---------------------------------------------------------------------------
__global__ __launch_bounds__(256) void rinv_kernel(const float* __restrict__ tok,
                                                   float* __restrict__ rinv) {
    int gt   = blockIdx.x * blockDim.x + threadIdx.x;
    int row  = gt >> 5;            // wave id == token row
    int lane = threadIdx.x & 31;
    const float* p = tok + (size_t)row * DIM + lane * 8;
    float4 a = *(const float4*)p;
    float4 b = *(const float4*)(p + 4);
    float s = a.x*a.x + a.y*a.y + a.z*a.z + a.w*a.w
            + b.x*b.x + b.y*b.y + b.z*b.z + b.w*b.w;
    #pragma unroll
    for (int off = 16; off > 0; off >>= 1) s += __shfl_xor(s, off, 32);
    if (lane == 0) rinv[row] = 1.0f / (sqrtf(s) + 1e-8f);
}

// ---------------------------------------------------------------------------
// Kernel 2: fused GEMM (f32 WMMA) + per-row top-8. One wave per 16-row tile.
// A = raw tokens (row scale is per-row constant -> doesn't change ordering),
// B = tokens scaled by rinv of the column token.
// ---------------------------------------------------------------------------
__global__ __launch_bounds__(32) void simtopk_kernel(const float* __restrict__ tok,
                                                     const float* __restrict__ rinv,
                                                     int* __restrict__ topk) {
    __shared__ float As[16][260];   // stride 260: conflict-free b64 frag loads
    __shared__ float Bs[16][260];
    __shared__ float Sm[16][17];    // 16x16 sim tile, padded

    const int lane = threadIdx.x;
    const int b    = blockIdx.x >> 6;       // 64 row-tiles per batch
    const int rt   = blockIdx.x & 63;
    const int r0   = rt * 16;

    const float* tb = tok  + (size_t)b * NTOK * DIM;
    const float* rb = rinv + (size_t)b * NTOK;

    // stage A tile: 16 rows x 256 f32 (1024 float4, 32 per lane)
    #pragma unroll 4
    for (int it = 0; it < 32; ++it) {
        int q   = lane + 32 * it;
        int row = q >> 6;
        int c4  = (q & 63) << 2;
        float4 v = *(const float4*)(tb + (size_t)(r0 + row) * DIM + c4);
        *(float4*)&As[row][c4] = v;
    }

    const int hi = lane >> 4;       // half-wave select (K pair)
    const int mn = lane & 15;       // M (A) / N (B,C) index
    const int grow = r0 + lane;     // global row owned by lanes 0..15

    float tv[KNN]; int ti[KNN];     // sorted descending top-8 (registers)
    #pragma unroll
    for (int s = 0; s < KNN; ++s) { tv[s] = -INFINITY; ti[s] = -1; }

    for (int ct = 0; ct < 64; ++ct) {
        const int c0 = ct * 16;
        // stage B tile, scaled by column rinv
        #pragma unroll 4
        for (int it = 0; it < 32; ++it) {
            int q   = lane + 32 * it;
            int row = q >> 6;
            int c4  = (q & 63) << 2;
            float sc = rb[c0 + row];
            float4 v = *(const float4*)(tb + (size_t)(c0 + row) * DIM + c4);
            v.x *= sc; v.y *= sc; v.z *= sc; v.w *= sc;
            *(float4*)&Bs[row][c4] = v;
        }
        __syncthreads();

        v8f c = {};
        #pragma unroll 4
        for (int k = 0; k < 64; ++k) {
            int kk = k * 4 + hi * 2;                  // {K, K+1} for this half
            v2f av = *(const v2f*)&As[mn][kk];
            v2f bv = *(const v2f*)&Bs[mn][kk];
            // D = A(16x4) x B(4x16) + C   -> v_wmma_f32_16x16x4_f32
            c = __builtin_amdgcn_wmma_f32_16x16x4_f32(
                    false, av, false, bv, (short)0, c, false, false);
        }

        // C layout: VGPR r -> M = r + 8*hi, N = mn
        #pragma unroll
        for (int r = 0; r < 8; ++r) Sm[r + hi * 8][mn] = c[r];
        __syncthreads();

        // lanes 0..15: merge 16 candidates of own row into sorted top-8
        if (lane < 16) {
            #pragma unroll 4
            for (int n = 0; n < 16; ++n) {
                int   gc = c0 + n;
                float v  = Sm[lane][n];
                if (gc != grow && v > tv[KNN - 1]) {
                    float cv = v; int ci = gc;          // branchless insert
                    #pragma unroll
                    for (int s = 0; s < KNN; ++s) {
                        bool  g  = cv > tv[s];
                        float nv = g ? cv : tv[s]; int ni = g ? ci : ti[s];
                        cv = g ? tv[s] : cv;       ci = g ? ti[s] : ci;
                        tv[s] = nv; ti[s] = ni;
                    }
                }
            }
        }
        __syncthreads();
    }

    if (lane < 16) {
        int* o = topk + ((size_t)b * NTOK + grow) * KNN;
        #pragma unroll
        for (int s = 0; s < KNN; ++s) o[s] = ti[s];
    }
}

// ---------------------------------------------------------------------------
// Kernel 3: zero the 128 MiB adjacency (the bandwidth-dominant step)
// ---------------------------------------------------------------------------
__global__ __launch_bounds__(256) void zero_kernel(float4* __restrict__ out, int n4) {
    int idx    = blockIdx.x * blockDim.x + threadIdx.x;
    int stride = gridDim.x * blockDim.x;
    float4 z = make_float4(0.f, 0.f, 0.f, 0.f);
    for (int i = idx; i < n4; i += stride) out[i] = z;
}

// ---------------------------------------------------------------------------
// Kernel 4: mutual check + scatter ones. out[b,i,j]=1 iff j in top(i) and
// i in top(j). One thread per row (8x8 integer compares).
// ---------------------------------------------------------------------------
__global__ __launch_bounds__(256) void edge_kernel(const int* __restrict__ topk,
                                                   float* __restrict__ out) {
    int i = blockIdx.x * blockDim.x + threadIdx.x;   // 0 .. B*N-1
    int b   = i >> 10;
    int row = i & (NTOK - 1);
    const int* mine = topk + (size_t)i * KNN;
    const int* tbat = topk + (size_t)b * NTOK * KNN;
    float* orow = out + (size_t)b * NTOK * NTOK + (size_t)row * NTOK;
    #pragma unroll
    for (int s = 0; s < KNN; ++s) {
        int j = mine[s];
        if (j < 0) continue;
        const int* theirs = tbat + (size_t)j * KNN;
        bool mutual = false;
        #pragma unroll
        for (int t = 0; t < KNN; ++t) mutual |= (theirs[t] == row);
        if (mutual) orow[j] = 1.0f;
    }
}

// ---------------------------------------------------------------------------
extern "C" void kernel_launch(void* const* d_in, const int* in_sizes, int n_in,
                              void* d_out, int out_size, void* d_ws, size_t ws_size,
                              hipStream_t stream) {
    const float* tok = (const float*)d_in[0];
    float* out = (float*)d_out;

    // ws layout: rinv (B*N f32 = 128 KiB) then topk (B*N*8 i32 = 1 MiB)
    float* rinv = (float*)d_ws;
    int*   topk = (int*)((char*)d_ws + (size_t)BATCH * NTOK * sizeof(float));

    // 1) reciprocal norms: 32768 waves, 8 waves/block
    rinv_kernel<<<(BATCH * NTOK) / 8, 256, 0, stream>>>(tok, rinv);

    // 2) fused WMMA similarity + top-8: one wave per 16-row tile
    simtopk_kernel<<<BATCH * (NTOK / 16), 32, 0, stream>>>(tok, rinv, topk);

    // 3) zero adjacency
    int n4 = (BATCH * NTOK * NTOK) / 4;
    zero_kernel<<<8192, 256, 0, stream>>>((float4*)out, n4);

    // 4) mutual edges
    edge_kernel<<<(BATCH * NTOK) / 256, 256, 0, stream>>>(topk, out);
}